// HaarWaveletTransform3D_68040871903411
// MI455X (gfx1250) — compile-verified
//
#include <hip/hip_runtime.h>

// Haar 3D wavelet transform, MI455X (gfx1250, wave32).
// x: (2,3,33,512,512) f32 -> out: (2,24,17,256,256) f32, out ch = filter*3 + ch.
// Pure streaming op (~420MB, zero reuse) => HBM-bound, floor ~18us @ 23.3TB/s.
// 8x8 Haar sign matrix runs on the WMMA engine (V_WMMA_F32_16X16X4_F32, two K=4
// steps accumulate the 8 block corners). One wave owns a full output row; the
// row id is readfirstlane()'d so all bases scalarize; the 8 filter-plane store
// bases are pinned into SGPR pairs via an empty asm so every store is pure
// SADDR + v(pos) + immediate (zero per-store address VALU).

typedef __attribute__((ext_vector_type(2))) float v2f;
typedef __attribute__((ext_vector_type(8))) float v8f;

#define HAAR_SCALE 0.3536f
#define NT 16  // w-tiles per output row (256 output cols / 16 per tile)
#define CH 4   // tiles per compute/store chunk (one EXEC toggle per chunk)

__global__ __launch_bounds__(256) void haar3d_wmma_kernel(const float* __restrict__ x,
                                                          float* __restrict__ out) {
  const int lane = threadIdx.x & 31;
  // Wave-uniform row id, made *provably* uniform so address math scalarizes.
  int id = __builtin_amdgcn_readfirstlane((int)(blockIdx.x * 8 + (threadIdx.x >> 5)));
  const int h  = id & 255; id >>= 8;              // 256 output rows
  const int t  = id % 17;                         // 17 output frames
  const int bc = id / 17;                         // 6 (b*c) slices
  const int b  = bc / 3;
  const int ch = bc - 3 * b;

  // causal pad: out frame t reads input frames (max(0,2t-1), 2t)
  const int t1 = 2 * t;
  const int t0 = (t1 == 0) ? 0 : (t1 - 1);

  // ---- A matrix: Haar signs * SCALE in 16x4 (rows 8..15 zero), per t-corner ----
  // A layout: lanes 0-15 -> M=lane, lanes 16-31 -> M=lane-16; VGPR0 K=2*half,
  // VGPR1 K=2*half+1.  K = tcorner*4 + hcorner*2 + wcorner.
  const int m  = lane & 15;
  const int Hk = lane >> 4;
  const float basev = (m < 8) ? HAAR_SCALE : 0.0f;
  const int wf = m & 1, hf = (m >> 1) & 1, tf = (m >> 2) & 1;
  const float a0 = (hf & Hk) ? -basev : basev;                // (h=Hk, w=0)
  const float a1 = ((wf ^ (hf & Hk)) != 0) ? -basev : basev;  // (h=Hk, w=1)
  v2f A0 = { a0, a1 };                          // t-corner 0
  v2f A1 = { tf ? -a0 : a0, tf ? -a1 : a1 };    // t-corner 1

  // ---- B loads: uniform SGPR frame bases + one per-lane 32-bit offset ----
  const unsigned HW    = 512u * 512u;
  const unsigned ubase = (unsigned)bc * 33u * HW + (unsigned)(2 * h) * 512u;  // uniform
  const float* p0 = x + ubase + (unsigned)t0 * HW;   // SGPR pair
  const float* p1 = x + ubase + (unsigned)t1 * HW;   // SGPR pair
  const int pos = lane & 15;
  const unsigned voff = (unsigned)(Hk * 512 + 2 * pos);  // per-lane, elements

  // preload whole row: 32 clause-able b64 NT loads, imm offset j*128B
  v2f B0[NT], B1[NT];
#pragma unroll
  for (int j = 0; j < NT; ++j) {
    B0[j] = __builtin_nontemporal_load(reinterpret_cast<const v2f*>(p0 + voff + j * 32));
    B1[j] = __builtin_nontemporal_load(reinterpret_cast<const v2f*>(p1 + voff + j * 32));
  }

  // ---- store bases: 8 uniform filter-plane pointers pinned in SGPR pairs ----
  const unsigned plane   = 256u * 256u;
  const unsigned fstride = 3u * 17u * plane;  // filter k -> k+1 channel stride (elements)
  const unsigned obase   = ((unsigned)(b * 24 + ch) * 17u + (unsigned)t) * plane
                         + (unsigned)h * 256u;  // uniform
  float* ob[8];
#pragma unroll
  for (int k = 0; k < 8; ++k) {
    float* pk = out + obase + (unsigned)k * fstride;
    asm("" : "+s"(pk));  // pin in SGPRs; blocks reassociation into VGPR addressing
    ob[k] = pk;
  }

#pragma unroll
  for (int jc = 0; jc < NT; jc += CH) {
    v8f c[CH];
#pragma unroll
    for (int u = 0; u < CH; ++u) {
      v8f z = {};
      // 8 args: (neg_a, A, neg_b, B, c_mod, C, reuse_a, reuse_b)
      z = __builtin_amdgcn_wmma_f32_16x16x4_f32(false, A0, false, B0[jc + u], (short)0, z, false, false);
      c[u] = __builtin_amdgcn_wmma_f32_16x16x4_f32(false, A1, false, B1[jc + u], (short)0, z, false, false);
    }
    // D: VGPR k, lanes 0-15 = filter k at the 16 positions (rows 8-15 are the
    // zero-padded half in lanes 16-31). SADDR stores: s[ob_k] + v(pos) + imm.
    if (lane < 16) {
#pragma unroll
      for (int u = 0; u < CH; ++u)
#pragma unroll
        for (int k = 0; k < 8; ++k)
          __builtin_nontemporal_store(c[u][k], ob[k] + (unsigned)pos + (unsigned)((jc + u) * 16));
    }
  }
}

extern "C" void kernel_launch(void* const* d_in, const int* in_sizes, int n_in,
                              void* d_out, int out_size, void* d_ws, size_t ws_size,
                              hipStream_t stream) {
  (void)in_sizes; (void)n_in; (void)d_ws; (void)ws_size; (void)out_size;
  const float* x = (const float*)d_in[0];  // (2,3,33,512,512) f32; d_in[1] (filters) baked in
  float* out = (float*)d_out;              // (2,24,17,256,256) f32

  // waves = 6 bc * 17 t * 256 h = 26112 (one per output row); 8 waves/block
  const int n_waves  = 6 * 17 * 256;
  const int n_blocks = n_waves / 8;        // 3264, exact
  haar3d_wmma_kernel<<<n_blocks, 256, 0, stream>>>(x, out);
}